// TopKTokenSampler_34857954574633
// MI455X (gfx1250) — compile-verified
//
#include <hip/hip_runtime.h>
#include <cstdint>

// ---------------------------------------------------------------------------
// Top-K(-inf)-masked softmax, B=256 rows x V=128000 vocab, K=50.
// Memory-bound (no matmul -> WMMA not applicable). CDNA5 features used:
//   * global_load_async_to_lds_b128 + s_wait_asynccnt (ASYNCcnt-tracked
//     async copy engine), 8-deep software pipeline for HBM latency hiding
//   * wave32 workgroup sizing (256 thr = 8 waves), LDS integer atomics
//   * non-temporal b128 output stores (write-once data; keep L2 for input)
// ---------------------------------------------------------------------------

#define VOCAB     128000
#define TOPK      50
#define THREADS   256
#define TILE_FL   (THREADS * 4)        // 1024 floats (4 KB) per pipeline step
#define NSTEPS    (VOCAB / TILE_FL)    // 125 (exact)
#define NBUF      8                    // async pipeline depth (7 outstanding)
#define NBINS     4096
#define NSUB      2
#define CAND_MAX  2048
#define NEG_BIG   (-3.402823466e38f)

typedef float v4f __attribute__((ext_vector_type(4)));  // native vector type

__device__ __forceinline__ unsigned fkey(float x) {
  // order-preserving float -> uint key (no NaNs in input)
  unsigned u = __float_as_uint(x);
  return (u & 0x80000000u) ? ~u : (u | 0x80000000u);
}

// async global -> LDS, 16 B per lane (GVS addressing: saddr + 32-bit vaddr)
__device__ __forceinline__ void async_issue(unsigned ldsAddr, unsigned byteOff,
                                            unsigned long long base) {
  asm volatile("global_load_async_to_lds_b128 %0, %1, %2"
               :: "v"(ldsAddr), "v"(byteOff), "s"(base)
               : "memory");
}

// wait until at most n async ops outstanding (clamped to NBUF-1)
__device__ __forceinline__ void async_wait_dyn(int n) {
  switch (n) {
    case 0: asm volatile("s_wait_asynccnt 0x0" ::: "memory"); break;
    case 1: asm volatile("s_wait_asynccnt 0x1" ::: "memory"); break;
    case 2: asm volatile("s_wait_asynccnt 0x2" ::: "memory"); break;
    case 3: asm volatile("s_wait_asynccnt 0x3" ::: "memory"); break;
    case 4: asm volatile("s_wait_asynccnt 0x4" ::: "memory"); break;
    case 5: asm volatile("s_wait_asynccnt 0x5" ::: "memory"); break;
    case 6: asm volatile("s_wait_asynccnt 0x6" ::: "memory"); break;
    default: asm volatile("s_wait_asynccnt 0x7" ::: "memory"); break;
  }
}

// Stream one 128000-float row through an NBUF-deep async global->LDS pipeline.
// Each thread consumes exactly the 16 B it issued -> no cross-thread LDS deps.
// Issue of step s+NBUF-1 recycles the buffer consumed at iteration s-1, whose
// ds_load values were already used -> write-after-read hazard is closed.
template <class F>
__device__ __forceinline__ void stream_row(float (*stage)[TILE_FL],
                                           unsigned ldsStage, int tid,
                                           unsigned long long rbase, F consume) {
  const unsigned slot = (unsigned)tid * 16u;
#pragma unroll
  for (int i = 0; i < NBUF - 1; ++i)
    async_issue(ldsStage + (unsigned)(i * TILE_FL * 4) + slot,
                (unsigned)(i * TILE_FL * 4) + slot, rbase);
  for (int s = 0; s < NSTEPS; ++s) {
    const int nxt = s + NBUF - 1;
    if (nxt < NSTEPS) {
      const int b = nxt & (NBUF - 1);
      async_issue(ldsStage + (unsigned)(b * TILE_FL * 4) + slot,
                  (unsigned)(nxt * TILE_FL * 4) + slot, rbase);
    }
    async_wait_dyn(NSTEPS - 1 - s);      // steady state: 0x7
    const v4f v = *(const v4f*)&stage[s & (NBUF - 1)][tid * 4];
    consume(s, v);
  }
}

__global__ __launch_bounds__(THREADS)
void topk_mask_softmax_kernel(const float* __restrict__ scores,
                              float* __restrict__ out) {
  const int row = blockIdx.x;
  const int tid = threadIdx.x;
  const float* rowp = scores + (size_t)row * VOCAB;
  float*       outp = out    + (size_t)row * VOCAB;
  const unsigned long long rbase = (unsigned long long)(uintptr_t)rowp;

  __shared__ __align__(16) float stage[NBUF][TILE_FL]; // 32 KB pipeline
  __shared__ int   hist[NSUB][NBINS];                  // 32 KB sub-histograms
  __shared__ float red[THREADS];                       // tree-reduce scratch
  __shared__ int   csum[THREADS];                      // per-chunk bin sums
  __shared__ float candVal[CAND_MAX];
  __shared__ int   candIdx[CAND_MAX];
  __shared__ int   selList[TOPK];
  __shared__ int   s_ncand, s_candBin, s_countAbove, s_selN;
  __shared__ float s_M, s_Sall, s_invS;

  const unsigned ldsStage = (unsigned)(uintptr_t)&stage[0][0];
  const int sub = (tid >> 4) & (NSUB - 1);  // split same-address LDS atomics

  for (int b = tid; b < NBINS; b += THREADS) { hist[0][b] = 0; hist[1][b] = 0; }
  if (tid == 0) s_ncand = 0;
  __syncthreads();

  // ---------------- Pass A: histogram + row max --------------------------
  float lmax = NEG_BIG;
  stream_row(stage, ldsStage, tid, rbase, [&](int s, v4f v) {
    const float xs[4] = {v.x, v.y, v.z, v.w};
#pragma unroll
    for (int j = 0; j < 4; ++j) {
      const float x = xs[j];
      lmax = fmaxf(lmax, x);
      atomicAdd(&hist[sub][fkey(x) >> 20], 1);
    }
  });
  red[tid] = lmax;
  __syncthreads();
  for (int off = THREADS / 2; off > 0; off >>= 1) {
    if (tid < off) red[tid] = fmaxf(red[tid], red[tid + off]);
    __syncthreads();
  }
  // merge sub-histograms into hist[0], then 16-bin chunk sums
  for (int b = tid; b < NBINS; b += THREADS) hist[0][b] += hist[1][b];
  __syncthreads();
  {
    int cs = 0;
#pragma unroll
    for (int j = 0; j < NBINS / THREADS; ++j)
      cs += hist[0][tid * (NBINS / THREADS) + j];
    csum[tid] = cs;
  }
  __syncthreads();
  if (tid == 0) {
    s_M = red[0];
    // find candidate bin: largest bin cb with count(bins >= cb) >= K
    int cum = 0, chunk = 0;
    for (int t = THREADS - 1; t >= 0; --t) {
      if (cum + csum[t] >= TOPK) { chunk = t; break; }
      cum += csum[t];
    }
    int cb = 0, above = 0;
    for (int b = chunk * (NBINS / THREADS) + (NBINS / THREADS) - 1;
         b >= chunk * (NBINS / THREADS); --b) {
      if (cum + hist[0][b] >= TOPK) { cb = b; above = cum; break; }
      cum += hist[0][b];
    }
    s_candBin = cb;
    s_countAbove = above;
  }
  __syncthreads();

  const float M  = s_M;
  const int   cb = s_candBin;

  // ---------------- Pass B: exp sums + candidate collection --------------
  float lall = 0.0f, ltop = 0.0f;
  stream_row(stage, ldsStage, tid, rbase, [&](int s, v4f v) {
    const float xs[4] = {v.x, v.y, v.z, v.w};
    const int gbase = s * TILE_FL + tid * 4;
#pragma unroll
    for (int j = 0; j < 4; ++j) {
      const float x = xs[j];
      const float e = __expf(x - M);
      lall += e;
      const int bin = (int)(fkey(x) >> 20);
      if (bin > cb) {
        ltop += e;
      } else if (bin == cb) {
        const int p = atomicAdd(&s_ncand, 1);
        if (p < CAND_MAX) { candVal[p] = x; candIdx[p] = gbase + j; }
      }
    }
  });
  red[tid] = lall;
  __syncthreads();
  for (int off = THREADS / 2; off > 0; off >>= 1) {
    if (tid < off) red[tid] += red[tid + off];
    __syncthreads();
  }
  if (tid == 0) s_Sall = red[0];
  __syncthreads();
  red[tid] = ltop;
  __syncthreads();
  for (int off = THREADS / 2; off > 0; off >>= 1) {
    if (tid < off) red[tid] += red[tid + off];
    __syncthreads();
  }
  if (tid == 0) {
    const float S_top = red[0];
    int n = s_ncand; if (n > CAND_MAX) n = CAND_MAX;
    int m = TOPK - s_countAbove; if (m > n) m = n;
    // exact select of the m largest candidates (value desc, index asc ties)
    float sumSel = 0.0f;
    int sn = 0;
    for (int jj = 0; jj < m; ++jj) {
      int best = -1, bi = 0x7fffffff;
      float bv = NEG_BIG;
      for (int i = 0; i < n; ++i) {
        const float cv = candVal[i];
        if (cv > bv || (cv == bv && candIdx[i] < bi)) {
          best = i; bv = cv; bi = candIdx[i];
        }
      }
      if (best < 0) break;
      sumSel += __expf(bv - M);
      selList[sn++] = candIdx[best];
      candVal[best] = NEG_BIG;   // remove from further consideration
    }
    s_selN = sn;
    const float S = s_Sall - S_top - sumSel;  // denominator over non-masked
    s_invS = 1.0f / S;
  }
  __syncthreads();

  const float invS = s_invS;
  const int   selN = s_selN;

  // ---------------- Pass C: write masked softmax (NT stores) -------------
  stream_row(stage, ldsStage, tid, rbase, [&](int s, v4f v) {
    const float xs[4] = {v.x, v.y, v.z, v.w};
    const int gbase = s * TILE_FL + tid * 4;
    float o[4];
#pragma unroll
    for (int j = 0; j < 4; ++j) {
      const float x = xs[j];
      const int bin = (int)(fkey(x) >> 20);
      bool masked = (bin > cb);
      if (bin == cb) {
        const int gi = gbase + j;
        for (int q = 0; q < selN; ++q)
          if (selList[q] == gi) { masked = true; break; }
      }
      o[j] = masked ? 0.0f : __expf(x - M) * invS;
    }
    v4f ov = {o[0], o[1], o[2], o[3]};
    __builtin_nontemporal_store(ov, (v4f*)(outp + gbase));
  });
}

extern "C" void kernel_launch(void* const* d_in, const int* in_sizes, int n_in,
                              void* d_out, int out_size, void* d_ws, size_t ws_size,
                              hipStream_t stream) {
  const float* scores = (const float*)d_in[0];
  float* out = (float*)d_out;
  const int B = in_sizes[0] / VOCAB;   // 256 rows
  topk_mask_softmax_kernel<<<dim3(B), dim3(THREADS), 0, stream>>>(scores, out);
}